// P3P_RANSAC_36335423324562
// MI455X (gfx1250) — compile-verified
//
#include <hip/hip_runtime.h>

#define N_PTS     8192
#define BATCH_HYP 2048
#define N_ITERS   2
#define NSAMP     6
#define INL_TH    4.0f
#define REG_EPS   1e-6f

typedef __attribute__((ext_vector_type(2))) float v2f;
typedef __attribute__((ext_vector_type(8))) float v8f;

// ---------------- Threefry-2x32 (bit-exact JAX PRNG) ----------------
__device__ __forceinline__ unsigned rotl32(unsigned x, int r) {
  return (x << r) | (x >> (32 - r));
}
#define TF_R(x0, x1, R) { x0 += x1; x1 = rotl32(x1, R); x1 ^= x0; }

__device__ __forceinline__ void threefry2x32(unsigned k0, unsigned k1,
                                             unsigned x0, unsigned x1,
                                             unsigned& o0, unsigned& o1) {
  unsigned k2 = k0 ^ k1 ^ 0x1BD11BDAu;
  x0 += k0; x1 += k1;
  TF_R(x0,x1,13) TF_R(x0,x1,15) TF_R(x0,x1,26) TF_R(x0,x1,6)
  x0 += k1; x1 += k2 + 1u;
  TF_R(x0,x1,17) TF_R(x0,x1,29) TF_R(x0,x1,16) TF_R(x0,x1,24)
  x0 += k2; x1 += k0 + 2u;
  TF_R(x0,x1,13) TF_R(x0,x1,15) TF_R(x0,x1,26) TF_R(x0,x1,6)
  x0 += k0; x1 += k1 + 3u;
  TF_R(x0,x1,17) TF_R(x0,x1,29) TF_R(x0,x1,16) TF_R(x0,x1,24)
  x0 += k1; x1 += k2 + 4u;
  TF_R(x0,x1,13) TF_R(x0,x1,15) TF_R(x0,x1,26) TF_R(x0,x1,6)
  x0 += k2; x1 += k0 + 5u;
  o0 = x0; o1 = x1;
}

__device__ __forceinline__ float bits_to_uniform(unsigned b) {
  unsigned f = (b >> 9) | 0x3F800000u;
  return __uint_as_float(f) - 1.0f;
}

// candidate (v,i) beats (v2,i2): larger value, ties -> lower index (lax.top_k)
__device__ __forceinline__ bool topk_better(float v, int i, float v2, int i2) {
  return (v > v2) || (v == v2 && i < i2);
}

// ---------------- Kernel 1: sample 6 points + DLT normal-equation solve ----
// grid = N_ITERS*BATCH_HYP blocks, 32 threads (1 wave) per model.
__global__ void k_sample_solve(const float* __restrict__ X,
                               const float* __restrict__ xi,
                               float* __restrict__ models) {
  const int g    = blockIdx.x;        // global model id
  const int iter = g >> 11;
  const int b    = g & (BATCH_HYP - 1);
  const int lane = threadIdx.x;

  // ki = fold_in(key(42), iter) = threefry((0,42),(0,iter))
  unsigned k0, k1;
  threefry2x32(0u, 42u, 0u, (unsigned)iter, k0, k1);

  // per-lane top-6 over a contiguous slice of 256 point indices
  float tv[NSAMP]; int ti[NSAMP];
  #pragma unroll
  for (int s = 0; s < NSAMP; ++s) { tv[s] = -__builtin_inff(); ti[s] = 0x7fffffff; }

  const unsigned half = 1024u * 8192u;   // half of 2048*8192 flat count
  for (int t = 0; t < 256; ++t) {
    int n = lane * 256 + t;
    unsigned j = (unsigned)b * 8192u + (unsigned)n;
    unsigned o0, o1; float val;
    if (b < 1024) { threefry2x32(k0, k1, j, j + half, o0, o1); val = bits_to_uniform(o0); }
    else          { threefry2x32(k0, k1, j - half, j, o0, o1); val = bits_to_uniform(o1); }
    if (topk_better(val, n, tv[NSAMP-1], ti[NSAMP-1])) {
      int s = NSAMP - 1;
      while (s > 0 && topk_better(val, n, tv[s-1], ti[s-1])) {
        tv[s] = tv[s-1]; ti[s] = ti[s-1]; --s;
      }
      tv[s] = val; ti[s] = n;
    }
  }

  __shared__ float sv[32 * NSAMP];
  __shared__ int   si[32 * NSAMP];
  #pragma unroll
  for (int s = 0; s < NSAMP; ++s) { sv[lane*NSAMP+s] = tv[s]; si[lane*NSAMP+s] = ti[s]; }
  __syncthreads();

  if (lane != 0) return;

  // merge 32*6 candidates -> final top-6 (exact lax.top_k order)
  float bv[NSAMP]; int bi[NSAMP];
  #pragma unroll
  for (int s = 0; s < NSAMP; ++s) { bv[s] = -__builtin_inff(); bi[s] = 0x7fffffff; }
  for (int c = 0; c < 32 * NSAMP; ++c) {
    float val = sv[c]; int n = si[c];
    if (topk_better(val, n, bv[NSAMP-1], bi[NSAMP-1])) {
      int s = NSAMP - 1;
      while (s > 0 && topk_better(val, n, bv[s-1], bi[s-1])) {
        bv[s] = bv[s-1]; bi[s] = bi[s-1]; --s;
      }
      bv[s] = val; bi[s] = n;
    }
  }

  // normal equations: AtA (11x11) + REG*I, Atb (11)
  float AtA[11][11], Atb[11];
  for (int i = 0; i < 11; ++i) { Atb[i] = 0.f; for (int j = 0; j < 11; ++j) AtA[i][j] = 0.f; }

  for (int s = 0; s < NSAMP; ++s) {
    int idx = bi[s];
    float Xs[4] = { X[idx*3+0], X[idx*3+1], X[idx*3+2], 1.0f };
    float u = xi[idx*2+0], v = xi[idx*2+1];
    float ru[12], rv[12];
    for (int k = 0; k < 4; ++k) {
      ru[k]   = Xs[k];  ru[4+k] = 0.f;    ru[8+k] = -u * Xs[k];
      rv[k]   = 0.f;    rv[4+k] = Xs[k];  rv[8+k] = -v * Xs[k];
    }
    for (int i = 0; i < 11; ++i) {
      for (int j = 0; j < 11; ++j) AtA[i][j] += ru[i]*ru[j] + rv[i]*rv[j];
      Atb[i] -= ru[i]*ru[11] + rv[i]*rv[11];
    }
  }
  for (int i = 0; i < 11; ++i) AtA[i][i] += REG_EPS;

  // Gaussian elimination w/ partial pivoting on augmented [AtA | Atb]
  float Aug[11][12];
  for (int i = 0; i < 11; ++i) { for (int j = 0; j < 11; ++j) Aug[i][j] = AtA[i][j]; Aug[i][11] = Atb[i]; }
  for (int col = 0; col < 11; ++col) {
    int piv = col; float pv = fabsf(Aug[col][col]);
    for (int r = col + 1; r < 11; ++r) {
      float a = fabsf(Aug[r][col]);
      if (a > pv) { pv = a; piv = r; }
    }
    if (piv != col)
      for (int c = col; c < 12; ++c) { float t = Aug[col][c]; Aug[col][c] = Aug[piv][c]; Aug[piv][c] = t; }
    float d = Aug[col][col];
    for (int r = col + 1; r < 11; ++r) {
      float f = Aug[r][col] / d;
      for (int c = col; c < 12; ++c) Aug[r][c] -= f * Aug[col][c];
    }
  }
  float p[11];
  for (int i = 10; i >= 0; --i) {
    float sacc = Aug[i][11];
    for (int j = i + 1; j < 11; ++j) sacc -= Aug[i][j] * p[j];
    p[i] = sacc / Aug[i][i];
  }
  for (int k = 0; k < 11; ++k) models[g*12 + k] = p[k];
  models[g*12 + 11] = 1.0f;
}

// ---------------- Kernel 2: WMMA scoring (branch-free, EXEC stays all-1s) --
// One wave scores 16 models. Per 16-point chunk: 3x v_wmma_f32_16x16x4_f32
// (D_r[p][m] = row-r projection of point p under model m); u,v,w lane-local.
__global__ void k_score(const float* __restrict__ X,
                        const float* __restrict__ xi,
                        const float* __restrict__ models,
                        float* __restrict__ scores) {
  const int tid  = blockIdx.x * blockDim.x + threadIdx.x;
  const int wave = tid >> 5;
  const int lane = tid & 31;
  const int n = lane & 15;     // model column
  const int h = lane >> 4;     // K/row half selector
  const int modelBase = wave * 16;

  // B_r: [k][n] = P_n[r, k]; VGPR j at lane (n,h) holds K = 2h + j
  const float* P = models + (modelBase + n) * 12;
  v2f b0, b1, b2;
  b0[0] = P[0*4 + 2*h]; b0[1] = P[0*4 + 2*h + 1];
  b1[0] = P[1*4 + 2*h]; b1[1] = P[1*4 + 2*h + 1];
  b2[0] = P[2*4 + 2*h]; b2[1] = P[2*4 + 2*h + 1];

  const float2* xi2 = (const float2*)xi;
  float acc = 0.0f;
  for (int c = 0; c < N_PTS / 16; ++c) {
    const int base = c * 16;
    // A: [m][k] = Xh[base+m, k]; lane (m,h) VGPR j holds K = 2h + j.
    // Branch-free: element 0 by computed address, element 1 load+select
    // (X[pA*3+1] is in-bounds for every lane; h==1 lanes replace it with 1).
    const int pA = base + n;
    v2f a;
    a[0] = X[pA*3 + 2*h];
    float ax1 = X[pA*3 + 1];
    a[1] = (h == 0) ? ax1 : 1.0f;

    v8f z = {0.f,0.f,0.f,0.f,0.f,0.f,0.f,0.f};
    v8f d0 = __builtin_amdgcn_wmma_f32_16x16x4_f32(false, a, false, b0, (short)0, z, false, false);
    v8f d1 = __builtin_amdgcn_wmma_f32_16x16x4_f32(false, a, false, b1, (short)0, z, false, false);
    v8f d2 = __builtin_amdgcn_wmma_f32_16x16x4_f32(false, a, false, b2, (short)0, z, false, false);

    // D layout: lane (n,h) holds rows M = v + 8h -> points base + 8h + v
    const int p0 = base + 8 * h;
    #pragma unroll
    for (int v = 0; v < 8; ++v) {
      const int pt = p0 + v;
      float2 uv = xi2[pt];                 // unconditional coalesced b64 load
      float w = d2[v];
      float zsafe = (w == 0.0f) ? -1.0f : w;   // reference z_safe semantics
      float du = d0[v] / zsafe - uv.x;         // exact IEEE division
      float dv = d1[v] / zsafe - uv.y;
      float s2 = fminf(du*du + dv*dv, INL_TH);
      acc += (w > 0.0f) ? s2 : INL_TH;         // chirality fail (0, NaN) -> 4
    }
  }
  acc += __shfl_xor(acc, 16, 32);
  if (h == 0) scores[modelBase + n] = acc;
}

// ---------------- Kernel 3: sequential-iteration argmin selection ----------
__device__ __forceinline__ bool score_less(float a, int ia, float b, int ib) {
  bool an = (a != a), bn = (b != b);
  if (an != bn) return an;        // NaN ranks as minimum (numpy argmin)
  if (an) return ia < ib;
  if (a != b) return a < b;
  return ia < ib;                 // first-index tie-break
}

__global__ void k_select(const float* __restrict__ scores,
                         const float* __restrict__ models,
                         float* __restrict__ out,
                         int* __restrict__ best_idx) {
  const int lane = threadIdx.x;   // single wave
  float best_score = __builtin_inff();
  int best = -1;
  for (int iter = 0; iter < N_ITERS; ++iter) {
    float s_min = __builtin_inff();
    int   i_min = 0x7fffffff;
    for (int k = lane; k < BATCH_HYP; k += 32) {
      float s = scores[iter*BATCH_HYP + k];
      if (score_less(s, k, s_min, i_min)) { s_min = s; i_min = k; }
    }
    for (int off = 16; off > 0; off >>= 1) {
      float os = __shfl_down(s_min, off, 32);
      int   oi = __shfl_down(i_min, off, 32);
      if (score_less(os, oi, s_min, i_min)) { s_min = os; i_min = oi; }
    }
    if (lane == 0 && s_min < best_score) {     // strict <, NaN -> keep old
      best_score = s_min;
      best = iter * BATCH_HYP + i_min;
    }
  }
  if (lane == 0) {
    best_idx[0] = best;
    for (int k = 0; k < 12; ++k)
      out[k] = (best >= 0) ? models[best*12 + k] : 0.0f;
  }
}

// ---------------- Kernel 4: inlier mask for winning model ------------------
__global__ void k_inliers(const float* __restrict__ X,
                          const float* __restrict__ xi,
                          const float* __restrict__ models,
                          const int* __restrict__ best_idx,
                          float* __restrict__ out) {
  const int p = blockIdx.x * blockDim.x + threadIdx.x;
  if (p >= N_PTS) return;
  const int bidx = best_idx[0];
  float r = 0.0f;
  if (bidx >= 0) {
    const float* P = models + bidx * 12;
    float X0 = X[p*3], X1 = X[p*3+1], X2 = X[p*3+2];
    float pr0 = P[0]*X0 + P[1]*X1 + P[2]*X2  + P[3];
    float pr1 = P[4]*X0 + P[5]*X1 + P[6]*X2  + P[7];
    float pr2 = P[8]*X0 + P[9]*X1 + P[10]*X2 + P[11];
    if (pr2 > 0.0f) {
      float du = pr0/pr2 - xi[p*2];
      float dv = pr1/pr2 - xi[p*2+1];
      if (du*du + dv*dv < INL_TH) r = 1.0f;
    }
  }
  out[12 + p] = r;
}

// ---------------- launch ---------------------------------------------------
extern "C" void kernel_launch(void* const* d_in, const int* in_sizes, int n_in,
                              void* d_out, int out_size, void* d_ws, size_t ws_size,
                              hipStream_t stream) {
  (void)in_sizes; (void)n_in; (void)out_size; (void)ws_size;
  const float* X  = (const float*)d_in[0];   // (8192,3)
  const float* xi = (const float*)d_in[1];   // (8192,2)
  float* out = (float*)d_out;                // 12 model floats + 8192 inlier flags

  float* ws      = (float*)d_ws;
  float* scores  = ws;                        // 4096 floats
  float* models  = ws + 4096;                 // 4096*12 floats
  int*   bestIdx = (int*)(ws + 4096 + 4096*12);

  k_sample_solve<<<N_ITERS * BATCH_HYP, 32, 0, stream>>>(X, xi, models);
  k_score<<<(N_ITERS * BATCH_HYP / 16) * 32 / 128, 128, 0, stream>>>(X, xi, models, scores);
  k_select<<<1, 32, 0, stream>>>(scores, models, out, bestIdx);
  k_inliers<<<N_PTS / 256, 256, 0, stream>>>(X, xi, models, bestIdx, out);
}